// Logic_53068615909594
// MI455X (gfx1250) — compile-verified
//
#include <hip/hip_runtime.h>
#include <cstdint>

// Problem geometry (from reference): X(8192,4096) f32, P(4,4096) f32, Y(8192,4096) f32.
// Collapsed math: Y[s,col] = P0[col] + x0*P1[col] + x1*P2[col] + x0*x1*P3[col],
// with (x0,x1) = X[s, 2b], X[s, 2b+1], col = 2b+o. Pure HBM-streaming kernel.

typedef __attribute__((ext_vector_type(4))) float v4f;

#define SROWS 8192
#define CCOLS 4096
#define TPB   256                 // 8 wave32 per workgroup
#define FPT   4                   // floats per thread per row (one b128)
#define COLS_PER_BLK (TPB * FPT)  // 1024 columns per block
#define GX (CCOLS / COLS_PER_BLK) // 4 blocks across columns
#define GY 256                    // row splits
#define ROWS_PER_BLK (SROWS / GY) // 32 rows per block

__global__ __launch_bounds__(TPB)
void logic_stream_kernel(const float* __restrict__ X,
                         const float* __restrict__ P,
                         float* __restrict__ Y) {
    // Double-buffered LDS staging: 2 buffers x 256 lanes x 16B = 8KB
    __shared__ v4f sh[2 * TPB];

    const int tid  = threadIdx.x;
    const int col  = blockIdx.x * COLS_PER_BLK + tid * FPT;
    const int row0 = blockIdx.y * ROWS_PER_BLK;

    // Per-column coefficients, loop-invariant -> registers (P sits in WGP$/L2)
    const v4f p0 = *(const v4f*)(P + 0 * CCOLS + col);
    const v4f p1 = *(const v4f*)(P + 1 * CCOLS + col);
    const v4f p2 = *(const v4f*)(P + 2 * CCOLS + col);
    const v4f p3 = *(const v4f*)(P + 3 * CCOLS + col);

    // LDS byte addresses for this lane's two slots. Generic (flat) pointers to
    // LDS carry the LDS byte offset in their low 32 bits on AMDGPU.
    const uint32_t lds_slot0 = (uint32_t)(uintptr_t)(&sh[tid]);
    const uint32_t lds_slot1 = (uint32_t)(uintptr_t)(&sh[TPB + tid]);

    // Prologue: async-copy row0's 16B for this lane into buffer 0 (ASYNCcnt++)
    {
        const float* g0 = X + (size_t)row0 * CCOLS + col;
        asm volatile("global_load_async_to_lds_b128 %0, %1, off"
                     :: "v"(lds_slot0), "v"(g0) : "memory");
    }

    int cur = 0;
    for (int r = 0; r < ROWS_PER_BLK; ++r) {
        const int row = row0 + r;

        // Prefetch next row into the other buffer (clamped on the last
        // iteration so control flow / ASYNCcnt bookkeeping stays uniform).
        const int nrow = (r + 1 < ROWS_PER_BLK) ? (row + 1) : row;
        const float*   gn   = X + (size_t)nrow * CCOLS + col;
        const uint32_t ldsn = (cur == 0) ? lds_slot1 : lds_slot0;
        asm volatile("global_load_async_to_lds_b128 %0, %1, off"
                     :: "v"(ldsn), "v"(gn) : "memory");

        // Async loads complete in order: allow exactly the just-issued copy to
        // remain outstanding -> current buffer is guaranteed resident in LDS.
        asm volatile("s_wait_asynccnt 0x1" ::: "memory");

        const v4f xv = sh[cur * TPB + tid];   // ds_load_b128, own slot only

        const float t01 = xv.x * xv.y;
        const float t23 = xv.z * xv.w;
        v4f y;
        y.x = fmaf(t01, p3.x, fmaf(xv.y, p2.x, fmaf(xv.x, p1.x, p0.x)));
        y.y = fmaf(t01, p3.y, fmaf(xv.y, p2.y, fmaf(xv.x, p1.y, p0.y)));
        y.z = fmaf(t23, p3.z, fmaf(xv.w, p2.z, fmaf(xv.z, p1.z, p0.z)));
        y.w = fmaf(t23, p3.w, fmaf(xv.w, p2.w, fmaf(xv.z, p1.w, p0.w)));

        // Non-temporal store: Y is write-once; keep the 128MB X (+64KB P)
        // resident in the 192MB L2 so graph replays re-read X from L2.
        __builtin_nontemporal_store(y, (v4f*)(Y + (size_t)row * CCOLS + col));

        cur ^= 1;
    }

    // Drain the final (redundant) prefetch explicitly.
    asm volatile("s_wait_asynccnt 0x0" ::: "memory");
}

extern "C" void kernel_launch(void* const* d_in, const int* in_sizes, int n_in,
                              void* d_out, int out_size, void* d_ws, size_t ws_size,
                              hipStream_t stream) {
    const float* X = (const float*)d_in[0];   // (8192, 4096) f32
    const float* P = (const float*)d_in[1];   // (4, 4096) f32
    float*       Y = (float*)d_out;           // (8192, 4096) f32

    dim3 grid(GX, GY);   // 4 x 256 = 1024 workgroups, 8 waves each
    dim3 block(TPB);
    logic_stream_kernel<<<grid, block, 0, stream>>>(X, P, Y);
}